// PAGATNet_88751204204885
// MI455X (gfx1250) — compile-verified
//
#include <hip/hip_runtime.h>

#define N_NODES   100000
#define EMB       64
#define HEADS     4
#define RD        32
#define HOUT      (HEADS * RD)   // 128
#define NEDGES    1600000
#define NEG_SLOPE 0.2f

typedef __attribute__((ext_vector_type(2))) float v2f;
typedef __attribute__((ext_vector_type(8))) float v8f;

// float atomic max via int max (val>=0) / uint min (val<0); valid with -inf init
__device__ __forceinline__ void atomicMaxF(float* addr, float v) {
    if (v >= 0.0f) atomicMax((int*)addr, __float_as_int(v));
    else           atomicMin((unsigned int*)addr, __float_as_uint(v));
}

__device__ __forceinline__ float leaky(float x) {
    return x >= 0.0f ? x : NEG_SLOPE * x;
}

// ---- init: zero out, m = -inf, denom = 0 (harness poisons buffers) ----
__global__ void init_kernel(float* __restrict__ out, float* __restrict__ m,
                            float* __restrict__ denom) {
    int i = blockIdx.x * blockDim.x + threadIdx.x;
    if (i < N_NODES * RD)    out[i] = 0.0f;
    if (i < N_NODES * HEADS) { m[i] = __int_as_float(0xFF800000); denom[i] = 0.0f; }
}

// ---- h = node_emb @ W via V_WMMA_F32_16X16X4_F32 ----
// 1 wave = 16-row tile x 128 cols; 16 k-steps x 8 col-tiles = 128 wmma/wave.
__global__ __launch_bounds__(256) void proj_wmma_kernel(const float* __restrict__ emb,
                                                        const float* __restrict__ W,
                                                        float* __restrict__ h) {
    __shared__ float Wt[HOUT][EMB + 1];   // W transposed, padded -> conflict-free ds_read_b64
    const int tid = threadIdx.x;
    for (int i = tid; i < EMB * HOUT; i += 256) {
        int k = i >> 7;        // row of W   (0..63)
        int n = i & 127;       // col of W   (0..127)
        Wt[n][k] = W[i];
    }
    __syncthreads();

    const int wave = tid >> 5;
    const int lane = tid & 31;
    const int rowTile = (blockIdx.x * 8 + wave) * 16;
    if (rowTile >= N_NODES) return;            // wave-uniform: EXEC stays all-ones

    const int half = lane >> 4;                // selects K pair {0,1} vs {2,3}
    const int lrow = lane & 15;                // M (for A) / N (for B,D)
    const float* arow = emb + (size_t)(rowTile + lrow) * EMB;

    v8f acc[8] = {};
    for (int k0 = 0; k0 < EMB; k0 += 4) {
        v2f a;
        a.x = arow[k0 + 2 * half];
        a.y = arow[k0 + 2 * half + 1];
        #pragma unroll
        for (int ct = 0; ct < 8; ++ct) {
            const int n = ct * 16 + lrow;
            v2f b;
            b.x = Wt[n][k0 + 2 * half];
            b.y = Wt[n][k0 + 2 * half + 1];
            acc[ct] = __builtin_amdgcn_wmma_f32_16x16x4_f32(
                false, a, false, b, (short)0, acc[ct], false, false);
        }
    }
    // D layout: VGPR v -> (M=v, N=lane) for lanes 0-15; (M=v+8, N=lane-16) for 16-31
    #pragma unroll
    for (int ct = 0; ct < 8; ++ct) {
        #pragma unroll
        for (int v = 0; v < 8; ++v) {
            const int mr = v + 8 * half;
            h[(size_t)(rowTile + mr) * HOUT + ct * 16 + lrow] = acc[ct][v];
        }
    }
}

// ---- per-node attention logits: a_src/a_dst [N, H] ----
__global__ void logits_kernel(const float* __restrict__ h,
                              const float* __restrict__ att_src,
                              const float* __restrict__ att_dst,
                              float* __restrict__ a_src, float* __restrict__ a_dst) {
    int i = blockIdx.x * blockDim.x + threadIdx.x;
    if (i >= N_NODES * HEADS) return;
    const int n = i >> 2, hh = i & 3;
    const float* hp = h + (size_t)n * HOUT + hh * RD;
    const float* as = att_src + hh * RD;
    const float* ad = att_dst + hh * RD;
    float s = 0.0f, d = 0.0f;
    #pragma unroll
    for (int k = 0; k < RD; ++k) { float v = hp[k]; s += v * as[k]; d += v * ad[k]; }
    a_src[i] = s;
    a_dst[i] = d;
}

// ---- pass A: per-edge logits + leaky + segment max ----
__global__ void edge_logit_max_kernel(const int* __restrict__ path,
                                      const float* __restrict__ a_src,
                                      const float* __restrict__ a_dst,
                                      float* __restrict__ ebuf, float* __restrict__ m) {
    int e = blockIdx.x * blockDim.x + threadIdx.x;
    if (e >= NEDGES) return;
    const int s = path[e], d = path[NEDGES + e];
    const float4 es = *(const float4*)(a_src + (size_t)s * 4);
    const float4 ed = *(const float4*)(a_dst + (size_t)d * 4);
    float4 ev;
    ev.x = leaky(es.x + ed.x);
    ev.y = leaky(es.y + ed.y);
    ev.z = leaky(es.z + ed.z);
    ev.w = leaky(es.w + ed.w);
    *(float4*)(ebuf + (size_t)e * 4) = ev;
    atomicMaxF(&m[d * 4 + 0], ev.x);
    atomicMaxF(&m[d * 4 + 1], ev.y);
    atomicMaxF(&m[d * 4 + 2], ev.z);
    atomicMaxF(&m[d * 4 + 3], ev.w);
}

// ---- pass B: ex = exp(e - m[dst]); denom += ex (ebuf overwritten with ex) ----
__global__ void edge_exp_sum_kernel(const int* __restrict__ path,
                                    float* __restrict__ ebuf,
                                    const float* __restrict__ m,
                                    float* __restrict__ denom) {
    int e = blockIdx.x * blockDim.x + threadIdx.x;
    if (e >= NEDGES) return;
    const int d = path[NEDGES + e];
    const float4 ev = *(const float4*)(ebuf + (size_t)e * 4);
    const float4 mv = *(const float4*)(m + (size_t)d * 4);
    float4 ex;
    ex.x = __expf(ev.x - mv.x);
    ex.y = __expf(ev.y - mv.y);
    ex.z = __expf(ev.z - mv.z);
    ex.w = __expf(ev.w - mv.w);
    *(float4*)(ebuf + (size_t)e * 4) = ex;
    atomicAdd(&denom[d * 4 + 0], ex.x);
    atomicAdd(&denom[d * 4 + 1], ex.y);
    atomicAdd(&denom[d * 4 + 2], ex.z);
    atomicAdd(&denom[d * 4 + 3], ex.w);
}

// ---- pass C: out[dst][d] += (1/H) * sum_h alpha[e,h] * h[src][h][d] ----
// one thread per (edge, d); head-mean folded in -> 1 atomic per output float
__global__ void aggregate_kernel(const int* __restrict__ path,
                                 const float* __restrict__ h,
                                 const float* __restrict__ ebuf,
                                 const float* __restrict__ denom,
                                 float* __restrict__ out) {
    int t = blockIdx.x * blockDim.x + threadIdx.x;
    if (t >= NEDGES * RD) return;
    const int e  = t >> 5;
    const int dd = t & 31;
    const int s = path[e], d = path[NEDGES + e];
    const float* hs = h + (size_t)s * HOUT;
    const float4 ex  = *(const float4*)(ebuf  + (size_t)e * 4);
    const float4 den = *(const float4*)(denom + (size_t)d * 4);
    float acc = (ex.x / (den.x + 1e-16f)) * hs[0 * RD + dd]
              + (ex.y / (den.y + 1e-16f)) * hs[1 * RD + dd]
              + (ex.z / (den.z + 1e-16f)) * hs[2 * RD + dd]
              + (ex.w / (den.w + 1e-16f)) * hs[3 * RD + dd];
    atomicAdd(&out[(size_t)d * RD + dd], 0.25f * acc);
}

extern "C" void kernel_launch(void* const* d_in, const int* in_sizes, int n_in,
                              void* d_out, int out_size, void* d_ws, size_t ws_size,
                              hipStream_t stream) {
    const float* node_emb = (const float*)d_in[0];
    const float* W        = (const float*)d_in[1];
    const float* att_src  = (const float*)d_in[2];
    const float* att_dst  = (const float*)d_in[3];
    const int*   path     = (const int*)d_in[4];
    float* out = (float*)d_out;

    // workspace layout (floats): total ~20.8M floats (~83 MB)
    float* ws    = (float*)d_ws;
    float* h     = ws;                                  // N * 128
    float* a_src = h     + (size_t)N_NODES * HOUT;      // N * 4
    float* a_dst = a_src + (size_t)N_NODES * HEADS;     // N * 4
    float* m     = a_dst + (size_t)N_NODES * HEADS;     // N * 4
    float* denom = m     + (size_t)N_NODES * HEADS;     // N * 4
    float* ebuf  = denom + (size_t)N_NODES * HEADS;     // E * 4

    init_kernel<<<(N_NODES * RD + 255) / 256, 256, 0, stream>>>(out, m, denom);

    const int rowTiles = N_NODES / 16;                  // 6250 exact
    proj_wmma_kernel<<<(rowTiles + 7) / 8, 256, 0, stream>>>(node_emb, W, h);

    logits_kernel<<<(N_NODES * HEADS + 255) / 256, 256, 0, stream>>>(
        h, att_src, att_dst, a_src, a_dst);

    edge_logit_max_kernel<<<(NEDGES + 255) / 256, 256, 0, stream>>>(
        path, a_src, a_dst, ebuf, m);

    edge_exp_sum_kernel<<<(NEDGES + 255) / 256, 256, 0, stream>>>(
        path, ebuf, m, denom);

    aggregate_kernel<<<(NEDGES * RD + 255) / 256, 256, 0, stream>>>(
        path, h, ebuf, denom, out);
}